// Robust_GSL_Block_54829552501122
// MI455X (gfx1250) — compile-verified
//
#include <hip/hip_runtime.h>
#include <hip/hip_bf16.h>

typedef float v2f __attribute__((ext_vector_type(2)));
typedef float v8f __attribute__((ext_vector_type(8)));

#define LRELU(v) fmaxf((v), 0.2f*(v))

__device__ __forceinline__ v8f wmma4(v2f a, v2f b, v8f c) {
  // D = A(16x4 f32) * B(4x16 f32) + C(16x16 f32)
  return __builtin_amdgcn_wmma_f32_16x16x4_f32(
      /*neg_a=*/false, a, /*neg_b=*/false, b,
      /*c_mod=*/(short)0, c, /*reuse_a=*/false, /*reuse_b=*/false);
}

__device__ __forceinline__ float wred_sum(float v) {
#pragma unroll
  for (int m = 16; m >= 1; m >>= 1) v += __shfl_xor(v, m, 32);
  return v;
}
__device__ __forceinline__ float wred_max(float v) {
#pragma unroll
  for (int m = 16; m >= 1; m >>= 1) v = fmaxf(v, __shfl_xor(v, m, 32));
  return v;
}

// B=8, C=128, N=4096, K=32, O=64, H=8
__global__ __launch_bounds__(256) void gsl_fused_kernel(
    const float* __restrict__ x,
    const float* __restrict__ w_trans, const float* __restrict__ g_trans, const float* __restrict__ b_trans,
    const float* __restrict__ w_gate1, const float* __restrict__ g_gate,  const float* __restrict__ b_gate,
    const float* __restrict__ w_gate2, const float* __restrict__ b_gate2,
    const float* __restrict__ w_attn1, const float* __restrict__ g_attn,  const float* __restrict__ b_attn,
    const float* __restrict__ w_attn2, const float* __restrict__ b_attn2,
    const float* __restrict__ agg_alpha,
    float* __restrict__ out_main, float* __restrict__ out_gate)
{
  constexpr int C = 128, KSTEPS = 32, NTILES = 5;
  constexpr size_t NK = 131072;        // N*K
  constexpr int GROUPS = 32768;        // B*N  (each group = one n, 32 k's)

  __shared__ float Bl[NTILES * KSTEPS * 64];   // 40 KB: weights in WMMA-B layout
  __shared__ float stage[8 * 64];              // per-wave logit staging

  // ---- build B-operand image: Bl[(kstep*5+t)*64 + 2*L + v] =
  //      W[o=16t + L%16][c = 4*kstep + v + 2*(L>>4)] ----
  for (int idx = threadIdx.x; idx < NTILES * KSTEPS * 64; idx += blockDim.x) {
    int v = idx & 1;
    int L = (idx >> 1) & 31;
    int blk = idx >> 6;
    int kstep = blk / NTILES;
    int t = blk - NTILES * kstep;
    int o16 = L & 15;
    int c = 4 * kstep + v + 2 * (L >> 4);
    float wv;
    if (t < 4)        wv = w_trans[(16 * t + o16) * C + c];
    else if (o16 < 8) wv = w_gate1[o16 * C + c];
    else              wv = w_attn1[(o16 - 8) * C + c];
    Bl[idx] = wv;
  }
  __syncthreads();

  const int lane  = threadIdx.x & 31;
  const int wave  = threadIdx.x >> 5;
  const int col   = lane & 15;       // WMMA N / C-matrix column
  const int halfB = lane >> 4;       // lane-half (row offset +8 in C layout)

  // per-lane affine params (column-indexed, matches C/D layout N=lane%16)
  float gT[4], bT[4];
#pragma unroll
  for (int t = 0; t < 4; ++t) { gT[t] = g_trans[16 * t + col]; bT[t] = b_trans[16 * t + col]; }
  float gH, bH, w2;
  if (col < 8) { gH = g_gate[col];     bH = b_gate[col];     w2 = w_gate2[col];     }
  else         { gH = g_attn[col - 8]; bH = b_attn[col - 8]; w2 = w_attn2[col - 8]; }
  const float bg2   = b_gate2[0];
  const float ba2   = b_attn2[0];
  const float alpha = 1.0f / (1.0f + __expf(-agg_alpha[0]));
  float* stg = stage + 64 * wave;

  const int gw     = blockIdx.x * 8 + wave;
  const int stride = gridDim.x * 8;
  const v8f vzero  = {0.f, 0.f, 0.f, 0.f, 0.f, 0.f, 0.f, 0.f};

  for (int g = gw; g < GROUPS; g += stride) {
    const int bb = g >> 12;               // batch (4096 groups per batch)
    const int n  = g & 4095;
    const size_t m0 = (size_t)g * 32;     // flat row base = b*NK + n*32

    // per-lane A base: row k = col (pair0) / col+16 (pair1); channels c, c+1 (half0) or c+2, c+3 (half1)
    const float* xa = x + ((size_t)bb * C + 2 * halfB) * NK + (size_t)n * 32 + col;
    __builtin_prefetch(xa, 0, 0);

    v8f acc[10];                          // [tile t][pair p] -> acc[2t+p]
#pragma unroll
    for (int i = 0; i < 10; ++i) acc[i] = vzero;

#pragma unroll 4
    for (int k = 0; k < KSTEPS; ++k) {
      const float* xk = xa + (size_t)(4 * k) * NK;
      v2f a0, a1;
      a0.x = xk[0];        a0.y = xk[NK];        // rows 0..15
      a1.x = xk[16];       a1.y = xk[NK + 16];   // rows 16..31
      const float* bp = &Bl[(k * NTILES) * 64 + 2 * lane];
#pragma unroll
      for (int t = 0; t < NTILES; ++t) {
        v2f bm = *(const v2f*)(bp + t * 64);     // ds_load_b64
        acc[2 * t + 0] = wmma4(a0, bm, acc[2 * t + 0]);
        acc[2 * t + 1] = wmma4(a1, bm, acc[2 * t + 1]);
      }
    }

    // ---- heads: affine+lrelu, 8->1 conv as butterfly over lane groups of 8 ----
    float hred[2][8];
#pragma unroll
    for (int p = 0; p < 2; ++p) {
#pragma unroll
      for (int j = 0; j < 8; ++j) {
        float v = LRELU(gH * acc[8 + p][j] + bH) * w2;
        v += __shfl_xor(v, 1, 32);
        v += __shfl_xor(v, 2, 32);
        v += __shfl_xor(v, 4, 32);
        hred[p][j] = v;   // lanes {0-7,16-23}: gate sum; lanes {8-15,24-31}: attn sum
      }
    }

    // stage per-row logits (row = j + 8*halfB + 16*p) so row <-> lane afterwards
#pragma unroll
    for (int p = 0; p < 2; ++p) {
#pragma unroll
      for (int j = 0; j < 8; ++j) {
        int row = j + 8 * halfB + 16 * p;
        if (col == 0) stg[row]      = hred[p][j] + bg2;
        if (col == 8) stg[32 + row] = hred[p][j] + ba2;
      }
    }
    asm volatile("s_wait_dscnt 0x0" ::: "memory");

    float gate_l = stg[lane];
    float attn_l = stg[32 + lane];
    out_gate[m0 + lane] = gate_l;                 // coalesced 128B store
    float msk = (gate_l >= 0.f) ? 1.f : 0.f;

    // softmax over K=32 (lane space), mask, renormalize
    float amax = wred_max(attn_l);
    float e  = __expf(attn_l - amax);
    float S  = wred_sum(e);
    float pm = (e / S) * msk;
    float Sm = wred_sum(pm);
    float wk = pm / (Sm + 1e-6f);

    // ---- weighted sum + unmasked max over the 32 rows for trans tiles ----
    float osum[4] = {0.f, 0.f, 0.f, 0.f};
    float omax[4] = {-3.4e38f, -3.4e38f, -3.4e38f, -3.4e38f};
#pragma unroll
    for (int p = 0; p < 2; ++p) {
#pragma unroll
      for (int j = 0; j < 8; ++j) {
        float wr = __shfl(wk, j + 8 * halfB + 16 * p, 32);  // ds_bpermute broadcast
#pragma unroll
        for (int t = 0; t < 4; ++t) {
          float tv = LRELU(gT[t] * acc[2 * t + p][j] + bT[t]);
          osum[t] = fmaf(wr, tv, osum[t]);
          omax[t] = fmaxf(omax[t], tv);
        }
      }
    }
#pragma unroll
    for (int t = 0; t < 4; ++t) {
      osum[t] += __shfl_xor(osum[t], 16, 32);
      omax[t]  = fmaxf(omax[t], __shfl_xor(omax[t], 16, 32));
      float r = alpha * osum[t] + (1.0f - alpha) * omax[t];
      if (lane < 16)
        out_main[((size_t)bb * 64 + 16 * t + lane) * 4096 + n] = r;
    }
  }
}

extern "C" void kernel_launch(void* const* d_in, const int* in_sizes, int n_in,
                              void* d_out, int out_size, void* d_ws, size_t ws_size,
                              hipStream_t stream) {
  (void)in_sizes; (void)n_in; (void)out_size; (void)d_ws; (void)ws_size;
  const float* x        = (const float*)d_in[0];
  const float* w_trans  = (const float*)d_in[1];
  const float* g_trans  = (const float*)d_in[2];
  const float* b_trans  = (const float*)d_in[3];
  const float* w_gate1  = (const float*)d_in[4];
  const float* g_gate   = (const float*)d_in[5];
  const float* b_gate   = (const float*)d_in[6];
  const float* w_gate2  = (const float*)d_in[7];
  const float* b_gate2  = (const float*)d_in[8];
  const float* w_attn1  = (const float*)d_in[9];
  const float* g_attn   = (const float*)d_in[10];
  const float* b_attn   = (const float*)d_in[11];
  const float* w_attn2  = (const float*)d_in[12];
  const float* b_attn2  = (const float*)d_in[13];
  const float* agg_a    = (const float*)d_in[14];

  float* out_main = (float*)d_out;
  float* out_gate = out_main + (size_t)8 * 64 * 4096;  // second tuple output

  dim3 grid(1024), block(256);
  hipLaunchKernelGGL(gsl_fused_kernel, grid, block, 0, stream,
                     x, w_trans, g_trans, b_trans,
                     w_gate1, g_gate, b_gate, w_gate2, b_gate2,
                     w_attn1, g_attn, b_attn, w_attn2, b_attn2,
                     agg_a, out_main, out_gate);
}